// contour_loss_module_25907242729908
// MI455X (gfx1250) — compile-verified
//
#include <hip/hip_runtime.h>
#include <hip/hip_bf16.h>
#include <math.h>

typedef __attribute__((ext_vector_type(2))) float v2f;
typedef __attribute__((ext_vector_type(8))) float v8f;
typedef __attribute__((ext_vector_type(4))) int   v4i;

#define B_    8
#define BASE_ 24
#define D0_   72
#define D1_   37
#define M_    (D0_ * D1_)   // 2664
#define N_    4096
#define MT_   256           // m-points per chamfer block
#define DEG5  0.08726646259971647f  // 5 degrees in radians

#define AS1 __attribute__((address_space(1)))
#define AS3 __attribute__((address_space(3)))

// ---------------------------------------------------------------------------
// Stage 1: rs = coef @ V via V_WMMA_F32_16X16X4_F32, then spherical->cartesian
// + clip, writing cv[B][M][3] to workspace.
// One wave (32 threads) per 16-column tile of V; 167 tiles.
// A (16x4 f32): lane -> M=lane%16; VGPR0 = K[0|2], VGPR1 = K[1|3] (lo/hi half).
// D (16x16 f32): lanes 0-15 hold N=lane, VGPR v = M-row v  -> each low lane
// holds radii for all 8 batches at its contour point.
// Boundary handling is branchless (clamped addresses + 0/1 weights) so EXEC
// stays all-ones and no exec-save branches surround the loads.
// ---------------------------------------------------------------------------
__global__ __launch_bounds__(32)
void cv_gemm_kernel(const float* __restrict__ coef,
                    const float* __restrict__ V,
                    const float* __restrict__ center,
                    float* __restrict__ cv) {
    const int lane  = threadIdx.x;
    const int tile  = blockIdx.x;
    const int row   = lane & 15;                 // A row (padded batch)
    const int rowc  = lane & 7;                  // clamped (valid) A row
    const int col   = tile * 16 + (lane & 15);   // V column / contour index m
    const int khalf = (lane < 16) ? 0 : 2;
    const bool cok  = (col < M_);
    const int  colc = cok ? col : (M_ - 1);      // clamped, always in-bounds
    const float wa  = (row < B_) ? 1.0f : 0.0f;  // zero out padded A rows
    const float wb  = cok ? 1.0f : 0.0f;         // zero out padded B cols

    // Uniform center loads (scalar-friendly), hoisted so they overlap WMMA.
    float ctr[3 * B_];
#pragma unroll
    for (int i = 0; i < 3 * B_; ++i) ctr[i] = center[i];

    v8f acc = {};
#pragma unroll
    for (int kk = 0; kk < 6; ++kk) {
        const int kb = kk * 4 + khalf;           // even -> float2 is 8B aligned
        const float2 av = *(const float2*)(coef + rowc * BASE_ + kb);
        v2f a, b;
        a.x = av.x * wa;
        a.y = av.y * wa;
        b.x = V[(kb + 0) * M_ + colc] * wb;
        b.y = V[(kb + 1) * M_ + colc] * wb;
        acc = __builtin_amdgcn_wmma_f32_16x16x4_f32(
            /*neg_a=*/false, a, /*neg_b=*/false, b,
            /*c_mod=*/(short)0, acc, /*reuse_a=*/false, /*reuse_b=*/false);
    }

    if (lane < 16 && cok) {
        const int i0 = col / D1_;        // phi index (step 5 deg)
        const int i1 = col - i0 * D1_;   // theta index (step 5 deg)
        float sph, cph, sth, cth;
        __sincosf((float)i0 * DEG5, &sph, &cph);
        __sincosf((float)i1 * DEG5, &sth, &cth);
#pragma unroll
        for (int b = 0; b < B_; ++b) {
            const float r  = acc[b];
            const float sx = r * sth * cph;
            const float sy = r * sth * sph;
            const float sz = r * cth;
            const float cx = ctr[b * 3 + 2] * 80.0f;
            const float cy = ctr[b * 3 + 1] * 80.0f;
            const float cz = ctr[b * 3 + 0] * 48.0f;
            const float z1 = fminf(fmaxf(sz + cz, 0.0f), 96.0f);
            const float y1 = fminf(fmaxf(sx + cy, 0.0f), 160.0f);
            const float x1 = fminf(fmaxf(sy + cx, 0.0f), 160.0f);
            float* o = cv + ((size_t)b * M_ + col) * 3;
            o[0] = z1; o[1] = y1; o[2] = x1;
        }
    }
}

// ---------------------------------------------------------------------------
// Stage 2: one-sided L1 chamfer. Block = (batch b, tile of 256 m-points).
// contour_gt[b] (48 KB) staged in LDS via async global->LDS b128 copies,
// each thread scans all 4096 GT points with b128 LDS loads (broadcast,
// conflict-free), then a deterministic block-sum of the per-point minima.
// ---------------------------------------------------------------------------
__global__ __launch_bounds__(MT_)
void chamfer_kernel(const float* __restrict__ cv,
                    const float* __restrict__ gt,
                    float* __restrict__ partial) {
    __shared__ float sgt[N_ * 3];   // 48 KB
    __shared__ float swr[MT_ / 32];

    const int b   = blockIdx.y;
    const int tid = threadIdx.x;
    const float* g = gt + (size_t)b * N_ * 3;

#if defined(__gfx1250__) && __has_builtin(__builtin_amdgcn_global_load_async_to_lds_b128) && __has_builtin(__builtin_amdgcn_s_wait_asynccnt)
    // 12288 floats = 3072 x b128; 256 threads -> 12 async b128 per thread.
    for (int i = tid * 4; i < N_ * 3; i += MT_ * 4) {
        __builtin_amdgcn_global_load_async_to_lds_b128(
            (v4i*)(g + i), (AS3 v4i*)(sgt + i), 0, 0);
    }
    __builtin_amdgcn_s_wait_asynccnt(0);
#else
    for (int i = tid * 4; i < N_ * 3; i += MT_ * 4) {
        *(float4*)(sgt + i) = *(const float4*)(g + i);
    }
#endif
    __syncthreads();

    const int m = blockIdx.x * MT_ + tid;
    float p0 = 0.0f, p1 = 0.0f, p2 = 0.0f;
    if (m < M_) {
        const float* c = cv + ((size_t)b * M_ + m) * 3;
        p0 = c[0]; p1 = c[1]; p2 = c[2];
    }

    float best = 3.402823466e38f;
    const float4* s4 = (const float4*)sgt;
#pragma unroll 2
    for (int n4 = 0; n4 < N_ / 4; ++n4) {
        const float4 qa = s4[3 * n4 + 0];
        const float4 qb = s4[3 * n4 + 1];
        const float4 qc = s4[3 * n4 + 2];
        const float d0 = fabsf(p0 - qa.x) + fabsf(p1 - qa.y) + fabsf(p2 - qa.z);
        const float d1 = fabsf(p0 - qa.w) + fabsf(p1 - qb.x) + fabsf(p2 - qb.y);
        const float d2 = fabsf(p0 - qb.z) + fabsf(p1 - qb.w) + fabsf(p2 - qc.x);
        const float d3 = fabsf(p0 - qc.y) + fabsf(p1 - qc.z) + fabsf(p2 - qc.w);
        best = fminf(best, fminf(fminf(d0, d1), fminf(d2, d3)));
    }
    if (m >= M_) best = 0.0f;

    // wave32 reduction, then cross-wave via LDS (deterministic order).
#pragma unroll
    for (int off = 16; off > 0; off >>= 1) best += __shfl_down(best, off);
    if ((tid & 31) == 0) swr[tid >> 5] = best;
    __syncthreads();
    if (tid == 0) {
        float s = 0.0f;
#pragma unroll
        for (int w = 0; w < MT_ / 32; ++w) s += swr[w];
        partial[blockIdx.y * gridDim.x + blockIdx.x] = s;
    }
}

// ---------------------------------------------------------------------------
// Stage 3: single-wave deterministic sum of block partials -> mean.
// ---------------------------------------------------------------------------
__global__ __launch_bounds__(32)
void finalize_kernel(const float* __restrict__ partial, int np,
                     float* __restrict__ out) {
    float s = 0.0f;
    for (int i = threadIdx.x; i < np; i += 32) s += partial[i];
#pragma unroll
    for (int off = 16; off > 0; off >>= 1) s += __shfl_down(s, off);
    if (threadIdx.x == 0) out[0] = s / (float)(B_ * M_);
}

extern "C" void kernel_launch(void* const* d_in, const int* in_sizes, int n_in,
                              void* d_out, int out_size, void* d_ws, size_t ws_size,
                              hipStream_t stream) {
    const float* coef   = (const float*)d_in[0];   // [8,24]
    const float* gt     = (const float*)d_in[1];   // [8,4096,3]
    const float* V      = (const float*)d_in[2];   // [24,2664]
    const float* center = (const float*)d_in[3];   // [8,3]
    float* out = (float*)d_out;

    float* cv      = (float*)d_ws;          // 8*2664*3 floats = 255.7 KB
    float* partial = cv + (size_t)B_ * M_ * 3;

    const int tiles  = (M_ + 15) / 16;      // 167
    cv_gemm_kernel<<<tiles, 32, 0, stream>>>(coef, V, center, cv);

    dim3 grid((M_ + MT_ - 1) / MT_, B_);    // 11 x 8
    chamfer_kernel<<<grid, MT_, 0, stream>>>(cv, gt, partial);

    finalize_kernel<<<1, 32, 0, stream>>>(partial, grid.x * B_, out);
}